// MedicalQueryGAT_25159918420788
// MI455X (gfx1250) — compile-verified
//
#include <hip/hip_runtime.h>
#include <hip/hip_bf16.h>
#include <math.h>

#define NNODES 50000
#define NEDGES 1600000
#define ETOT   (NEDGES + NNODES)
#define NGRAPH 512
#define INDIM  128
#define HIDC   32
#define HEADS  4
#define D1     (HEADS * HIDC)   // 128

typedef float v2f __attribute__((ext_vector_type(2)));
typedef float v8f __attribute__((ext_vector_type(8)));

static inline int cdiv(long long a, long long b) { return (int)((a + b - 1) / b); }

// -------------------- init --------------------
__global__ void k_fill(float* __restrict__ p, float v, int n) {
    int i = blockIdx.x * blockDim.x + threadIdx.x;
    if (i < n) p[i] = v;
}

// -------------------- fp32 WMMA GEMM: C[M,N] = A[M,K] @ B[K,N] --------------------
// Each wave computes MT vertically-stacked 16x16 tiles (same 16 columns), so one
// B fragment feeds MT consecutive v_wmma_f32_16x16x4_f32 ops (5x less B traffic).
// blockDim.x = 32 * (N/16); gridDim.x = M / (16*MT). EXEC all-ones (WMMA requirement).
template <int K, int N, int MT>
__global__ void k_gemm_wmma_f32(const float* __restrict__ A,
                                const float* __restrict__ B,
                                float* __restrict__ C) {
    const int lane = threadIdx.x & 31;
    const int wave = threadIdx.x >> 5;
    const int half = lane >> 4;          // 0: lanes 0-15, 1: lanes 16-31
    const int l    = lane & 15;
    const int m0   = blockIdx.x * (16 * MT);
    const int n0   = wave * 16;

    v8f acc[MT];
    #pragma unroll
    for (int t = 0; t < MT; ++t) acc[t] = (v8f){};

    for (int kk = 0; kk < K; kk += 4) {
        const int ka = kk + 2 * half;    // vgpr r of A/B holds K = 2*half + r
        v2f b;                           // 4x16 B fragment, shared by all MT tiles
        b.x = B[(size_t)ka * N + n0 + l];
        b.y = B[(size_t)(ka + 1) * N + n0 + l];
        #pragma unroll
        for (int t = 0; t < MT; ++t) {
            v2f a;                       // contiguous pair -> b64 load
            const float* arow = A + (size_t)(m0 + t * 16 + l) * K;
            a.x = arow[ka];
            a.y = arow[ka + 1];
            acc[t] = __builtin_amdgcn_wmma_f32_16x16x4_f32(
                false, a, false, b, (short)0, acc[t], false, false);
        }
    }

    #pragma unroll
    for (int t = 0; t < MT; ++t) {
        #pragma unroll
        for (int r = 0; r < 8; ++r) {
            const int row = m0 + t * 16 + r + 8 * half;  // C: vgpr r -> M=r / M=r+8 by lane half
            C[(size_t)row * N + n0 + l] = acc[t][r];
        }
    }
}

// -------------------- per-node attention coefficients --------------------
__global__ void k_alpha1(const float* __restrict__ H1,
                         const float* __restrict__ aw_s, const float* __restrict__ aw_d,
                         float* __restrict__ as_o, float* __restrict__ ad_o) {
    int id = blockIdx.x * blockDim.x + threadIdx.x;
    if (id >= NNODES * HEADS) return;
    int n = id >> 2, hd = id & 3;
    const float* hp = H1 + (size_t)n * D1 + hd * HIDC;
    const float* sp = aw_s + hd * HIDC;
    const float* dp = aw_d + hd * HIDC;
    float s = 0.f, d = 0.f;
    #pragma unroll
    for (int c = 0; c < HIDC; ++c) { float v = hp[c]; s += v * sp[c]; d += v * dp[c]; }
    as_o[id] = s; ad_o[id] = d;
}

__global__ void k_alpha2(const float* __restrict__ H2,
                         const float* __restrict__ aw_s, const float* __restrict__ aw_d,
                         float* __restrict__ as_o, float* __restrict__ ad_o) {
    int n = blockIdx.x * blockDim.x + threadIdx.x;
    if (n >= NNODES) return;
    const float* hp = H2 + (size_t)n * HIDC;
    float s = 0.f, d = 0.f;
    #pragma unroll
    for (int c = 0; c < HIDC; ++c) { float v = hp[c]; s += v * aw_s[c]; d += v * aw_d[c]; }
    as_o[n] = s; ad_o[n] = d;
}

// -------------------- edge helpers --------------------
__device__ __forceinline__ void edge_nodes(const int* __restrict__ ei, int i, int& s, int& d) {
    if (i < NEDGES) { s = ei[i]; d = ei[NEDGES + i]; }
    else            { s = i - NEDGES; d = s; }          // self loops
}

__device__ __forceinline__ float lrelu(float v) { return v > 0.f ? v : 0.2f * v; }

__device__ __forceinline__ void atomicMaxF(float* a, float v) {
    // lowers to global_atomic_max_num_f32 (device scope, relaxed)
    __hip_atomic_fetch_max(a, v, __ATOMIC_RELAXED, __HIP_MEMORY_SCOPE_AGENT);
}

// segment max of leaky-relu'd logits into m[dst*H+h]
template <int HH>
__global__ void k_edge_max(const int* __restrict__ ei,
                           const float* __restrict__ as_, const float* __restrict__ ad_,
                           float* __restrict__ m) {
    int id = blockIdx.x * blockDim.x + threadIdx.x;
    if (id >= ETOT * HH) return;
    int e = id / HH, hd = id % HH;
    int s, d; edge_nodes(ei, e, s, d);
    float v = lrelu(as_[s * HH + hd] + ad_[d * HH + hd]);
    atomicMaxF(&m[d * HH + hd], v);
}

// segment sum of exp(e - m) into den[dst*H+h]
template <int HH>
__global__ void k_edge_den(const int* __restrict__ ei,
                           const float* __restrict__ as_, const float* __restrict__ ad_,
                           const float* __restrict__ m, float* __restrict__ den) {
    int id = blockIdx.x * blockDim.x + threadIdx.x;
    if (id >= ETOT * HH) return;
    int e = id / HH, hd = id % HH;
    int s, d; edge_nodes(ei, e, s, d);
    float v = lrelu(as_[s * HH + hd] + ad_[d * HH + hd]);
    atomicAdd(&den[d * HH + hd], __expf(v - m[d * HH + hd]));
}

// layer-1 aggregation: one block per edge, 128 threads = 4 heads x 32 ch.
// H1 (25.6MB) and out1 (25.6MB) are L2-resident -> gather + atomic scatter stay in L2.
__global__ void k_edge_agg1(const int* __restrict__ ei,
                            const float* __restrict__ as_, const float* __restrict__ ad_,
                            const float* __restrict__ m, const float* __restrict__ den,
                            const float* __restrict__ H1, float* __restrict__ out) {
    int e = blockIdx.x;
    int c = threadIdx.x;            // 0..127
    int hd = c >> 5;
    int s, d; edge_nodes(ei, e, s, d);
    float v = lrelu(as_[s * 4 + hd] + ad_[d * 4 + hd]);
    float w = __expf(v - m[d * 4 + hd]) / den[d * 4 + hd];
    atomicAdd(&out[(size_t)d * D1 + c], w * H1[(size_t)s * D1 + c]);
}

// layer-2 aggregation: 32 ch per edge, 8 edges per 256-thread block.
__global__ void k_edge_agg2(const int* __restrict__ ei,
                            const float* __restrict__ as_, const float* __restrict__ ad_,
                            const float* __restrict__ m, const float* __restrict__ den,
                            const float* __restrict__ H2, float* __restrict__ out) {
    int id = blockIdx.x * blockDim.x + threadIdx.x;
    int e = id >> 5;
    if (e >= ETOT) return;
    int c = id & 31;
    int s, d; edge_nodes(ei, e, s, d);
    float v = lrelu(as_[s] + ad_[d]);
    float w = __expf(v - m[d]) / den[d];
    atomicAdd(&out[(size_t)d * HIDC + c], w * H2[(size_t)s * HIDC + c]);
}

// -------------------- bias + ELU (in place) --------------------
template <int NCOLS>
__global__ void k_bias_elu(float* __restrict__ x, const float* __restrict__ b, int total) {
    int i = blockIdx.x * blockDim.x + threadIdx.x;
    if (i >= total) return;
    float v = x[i] + b[i & (NCOLS - 1)];   // NCOLS is a power of two
    x[i] = v > 0.f ? v : (__expf(v) - 1.f);
}

// -------------------- global mean pool (sum + count) --------------------
__global__ void k_pool(const float* __restrict__ h, const int* __restrict__ batch,
                       float* __restrict__ pooled, float* __restrict__ cnt) {
    int i = blockIdx.x * blockDim.x + threadIdx.x;
    if (i >= NNODES * HIDC) return;
    int n = i >> 5, c = i & 31;
    int g = batch[n];
    atomicAdd(&pooled[g * HIDC + c], h[i]);
    if (c == 0) atomicAdd(&cnt[g], 1.0f);
}

// -------------------- classifier: 32 -> 16 (relu) -> 1 --------------------
__global__ void k_classifier(const float* __restrict__ pooled, const float* __restrict__ cnt,
                             const float* __restrict__ Wc1, const float* __restrict__ bc1,
                             const float* __restrict__ Wc2, const float* __restrict__ bc2,
                             float* __restrict__ out) {
    int g = blockIdx.x * blockDim.x + threadIdx.x;
    if (g >= NGRAPH) return;
    float inv = 1.0f / cnt[g];
    float p[HIDC];
    #pragma unroll
    for (int c = 0; c < HIDC; ++c) p[c] = pooled[g * HIDC + c] * inv;
    float acc = bc2[0];
    #pragma unroll
    for (int j = 0; j < HIDC / 2; ++j) {
        float z = bc1[j];
        #pragma unroll
        for (int c = 0; c < HIDC; ++c) z += p[c] * Wc1[c * (HIDC / 2) + j];
        z = z > 0.f ? z : 0.f;
        acc += z * Wc2[j];
    }
    out[g] = acc;
}

// -------------------- launch --------------------
extern "C" void kernel_launch(void* const* d_in, const int* in_sizes, int n_in,
                              void* d_out, int out_size, void* d_ws, size_t ws_size,
                              hipStream_t stream) {
    const float* x     = (const float*)d_in[0];
    const int*   ei    = (const int*)d_in[1];
    // d_in[2] edge_attr: unused by the reference math
    const int*   batch = (const int*)d_in[3];
    const float* W1    = (const float*)d_in[4];
    const float* aw_s1 = (const float*)d_in[5];
    const float* aw_d1 = (const float*)d_in[6];
    const float* b1    = (const float*)d_in[7];
    const float* W2    = (const float*)d_in[8];
    const float* aw_s2 = (const float*)d_in[9];
    const float* aw_d2 = (const float*)d_in[10];
    const float* b2    = (const float*)d_in[11];
    const float* Wc1   = (const float*)d_in[12];
    const float* bc1   = (const float*)d_in[13];
    const float* Wc2   = (const float*)d_in[14];
    const float* bc2   = (const float*)d_in[15];
    float* out = (float*)d_out;

    // workspace layout (floats); total ~17.0M floats (~68 MB) -> fits 192MB L2
    float* ws = (float*)d_ws;
    size_t o = 0;
    float* H1   = ws + o; o += (size_t)NNODES * D1;    // 6.4M
    float* out1 = ws + o; o += (size_t)NNODES * D1;    // 6.4M (becomes ELU'd input of layer 2)
    float* H2   = ws + o; o += (size_t)NNODES * HIDC;  // 1.6M
    float* out2 = ws + o; o += (size_t)NNODES * HIDC;  // 1.6M
    float* as1  = ws + o; o += (size_t)NNODES * HEADS;
    float* ad1  = ws + o; o += (size_t)NNODES * HEADS;
    float* m1   = ws + o; o += (size_t)NNODES * HEADS;
    float* den1 = ws + o; o += (size_t)NNODES * HEADS;
    float* as2  = ws + o; o += (size_t)NNODES;
    float* ad2  = ws + o; o += (size_t)NNODES;
    float* m2   = ws + o; o += (size_t)NNODES;
    float* den2 = ws + o; o += (size_t)NNODES;
    float* pooled = ws + o; o += (size_t)NGRAPH * HIDC;
    float* cnt    = ws + o; o += (size_t)NGRAPH;

    const float NEG_INF = -__builtin_inff();
    const int MT = 5;                       // 3125 M-tiles = 5 * 625 exactly

    // ---- layer 1 ----
    k_gemm_wmma_f32<INDIM, D1, MT><<<NNODES / (16 * MT), 8 * 32, 0, stream>>>(x, W1, H1);
    k_alpha1<<<cdiv((long long)NNODES * HEADS, 256), 256, 0, stream>>>(H1, aw_s1, aw_d1, as1, ad1);
    k_fill<<<cdiv((long long)NNODES * HEADS, 256), 256, 0, stream>>>(m1, NEG_INF, NNODES * HEADS);
    k_fill<<<cdiv((long long)NNODES * HEADS, 256), 256, 0, stream>>>(den1, 0.f, NNODES * HEADS);
    k_fill<<<cdiv((long long)NNODES * D1, 256), 256, 0, stream>>>(out1, 0.f, NNODES * D1);
    k_edge_max<HEADS><<<cdiv((long long)ETOT * HEADS, 256), 256, 0, stream>>>(ei, as1, ad1, m1);
    k_edge_den<HEADS><<<cdiv((long long)ETOT * HEADS, 256), 256, 0, stream>>>(ei, as1, ad1, m1, den1);
    k_edge_agg1<<<ETOT, D1, 0, stream>>>(ei, as1, ad1, m1, den1, H1, out1);
    k_bias_elu<D1><<<cdiv((long long)NNODES * D1, 256), 256, 0, stream>>>(out1, b1, NNODES * D1);

    // ---- layer 2 ----
    k_gemm_wmma_f32<D1, HIDC, MT><<<NNODES / (16 * MT), 2 * 32, 0, stream>>>(out1, W2, H2);
    k_alpha2<<<cdiv(NNODES, 256), 256, 0, stream>>>(H2, aw_s2, aw_d2, as2, ad2);
    k_fill<<<cdiv(NNODES, 256), 256, 0, stream>>>(m2, NEG_INF, NNODES);
    k_fill<<<cdiv(NNODES, 256), 256, 0, stream>>>(den2, 0.f, NNODES);
    k_fill<<<cdiv((long long)NNODES * HIDC, 256), 256, 0, stream>>>(out2, 0.f, NNODES * HIDC);
    k_edge_max<1><<<cdiv(ETOT, 256), 256, 0, stream>>>(ei, as2, ad2, m2);
    k_edge_den<1><<<cdiv(ETOT, 256), 256, 0, stream>>>(ei, as2, ad2, m2, den2);
    k_edge_agg2<<<cdiv((long long)ETOT * HIDC, 256), 256, 0, stream>>>(ei, as2, ad2, m2, den2, H2, out2);
    k_bias_elu<HIDC><<<cdiv((long long)NNODES * HIDC, 256), 256, 0, stream>>>(out2, b2, NNODES * HIDC);

    // ---- pool + classifier ----
    k_fill<<<cdiv(NGRAPH * HIDC + NGRAPH, 256), 256, 0, stream>>>(pooled, 0.f, NGRAPH * HIDC + NGRAPH); // pooled + cnt contiguous
    k_pool<<<cdiv((long long)NNODES * HIDC, 256), 256, 0, stream>>>(out2, batch, pooled, cnt);
    k_classifier<<<cdiv(NGRAPH, 256), 256, 0, stream>>>(pooled, cnt, Wc1, bc1, Wc2, bc2, out);
}